// LSTMModel_50757923504889
// MI455X (gfx1250) — compile-verified
//
#include <hip/hip_runtime.h>
#include <hip/hip_bf16.h>

// LSTM: T=4096, IN=256, H=1024, L=4, OUT=10, batch=1.
// Phase 1 (per layer): xz = X @ W_ih^T + b   -> fp32 WMMA GEMM (parallel in T)
// Phase 2 (per layer): sequential scan, z_t = xz_t + W_hh @ h_{t-1} -> persistent
//                      kernel, VALU matvec (N=1), software grid barrier per step.
// Phase 3: out = h @ fc_W^T + fc_b (tiny).

typedef __attribute__((ext_vector_type(2))) float v2f;
typedef __attribute__((ext_vector_type(8))) float v8f;

#define T_SEQ   4096
#define HDIM    1024
#define G4H     4096   // 4*H
#define RBLOCKS 64     // blocks in recurrent kernel (j-tiles of 16)

// ---------------------------------------------------------------- utilities
__device__ __forceinline__ float sigmoidf_(float x) {
    return 1.0f / (1.0f + __expf(-x));
}

// Sense-reversal grid barrier. bar[0]=arrive count, bar[1]=generation.
__device__ __forceinline__ void grid_barrier(unsigned* bar, unsigned nblk) {
    __syncthreads();
    if (threadIdx.x == 0) {
        __threadfence();
        unsigned gen = __hip_atomic_load(&bar[1], __ATOMIC_RELAXED,
                                         __HIP_MEMORY_SCOPE_AGENT);
        unsigned arrived = __hip_atomic_fetch_add(&bar[0], 1u, __ATOMIC_ACQ_REL,
                                                  __HIP_MEMORY_SCOPE_AGENT);
        if (arrived == nblk - 1u) {
            __hip_atomic_store(&bar[0], 0u, __ATOMIC_RELAXED,
                               __HIP_MEMORY_SCOPE_AGENT);
            __hip_atomic_fetch_add(&bar[1], 1u, __ATOMIC_RELEASE,
                                   __HIP_MEMORY_SCOPE_AGENT);
        } else {
            while (__hip_atomic_load(&bar[1], __ATOMIC_ACQUIRE,
                                     __HIP_MEMORY_SCOPE_AGENT) == gen) {
                __builtin_amdgcn_s_sleep(2);
            }
        }
        __threadfence();
    }
    __syncthreads();
}

__global__ void init_barrier_kernel(unsigned* bar) {
    if (threadIdx.x < 2) bar[threadIdx.x] = 0u;
}

// ---------------------------------------------------------- WMMA fp32 GEMM
// XZ[M=4096, N=4096] = X[M, Din] @ W[N, Din]^T + bias[N]
// One wave -> one 16x16 tile, K stepped by 4 with V_WMMA_F32_16X16X4_F32.
// A lane layout (16x4 f32): lanes 0-15 hold M=lane, K={0,1}; lanes 16-31 K={2,3}.
// B lane layout (4x16 f32): lanes 0-15 hold N=lane, K={0,1}; lanes 16-31 K={2,3}.
// C/D: VGPR r, lane<16 -> (M=r, N=lane); lane>=16 -> (M=r+8, N=lane-16).
__global__ __launch_bounds__(256) void gemm_xz_kernel(
        const float* __restrict__ X, const float* __restrict__ W,
        const float* __restrict__ bias, float* __restrict__ XZ, int Din) {
    const int lane = threadIdx.x & 31;
    const int wave = threadIdx.x >> 5;              // 8 waves per block
    const int n0   = blockIdx.x * 16;               // 256 n-tiles
    const int m0   = (blockIdx.y * 8 + wave) * 16;  // 32*8 = 256 m-tiles
    const int l16  = lane & 15;
    const int kh   = lane >> 4;                     // 0 or 1 (K half)

    const float* aptr = X + (size_t)(m0 + l16) * Din + 2 * kh;
    const float* bptr = W + (size_t)(n0 + l16) * Din + 2 * kh;

    v8f acc = {0.f, 0.f, 0.f, 0.f, 0.f, 0.f, 0.f, 0.f};

#if __has_builtin(__builtin_amdgcn_wmma_f32_16x16x4_f32)
    #pragma unroll 4
    for (int k = 0; k < Din; k += 4) {
        v2f a = *(const v2f*)(aptr + k);
        v2f b = *(const v2f*)(bptr + k);
        // (neg_a, A, neg_b, B, c_mod, C, reuse_a, reuse_b)
        acc = __builtin_amdgcn_wmma_f32_16x16x4_f32(
            false, a, false, b, (short)0, acc, false, false);
    }
#else
    // Scalar fallback (not expected on gfx1250).
    for (int r = 0; r < 8; ++r) {
        const float* xr = X + (size_t)(m0 + kh * 8 + r) * Din;
        const float* wr = W + (size_t)(n0 + l16) * Din;
        float s = 0.f;
        for (int k = 0; k < Din; ++k) s = fmaf(xr[k], wr[k], s);
        acc[r] = s;
    }
#endif

    const float bn = bias[n0 + l16];
    float* op = XZ + (size_t)(m0 + kh * 8) * G4H + n0 + l16;
    #pragma unroll
    for (int r = 0; r < 8; ++r)
        op[(size_t)r * G4H] = acc[r] + bn;
}

// ------------------------------------------------------- recurrent scan
// Persistent kernel: 64 blocks x 256 threads. Block b owns hidden units
// j in [16b, 16b+16). Each quad of lanes computes one of the 64 gate rows
// (4 gates x 16 units) as a 1024-length dot against LDS-staged h_{t-1}.
// c lives in registers of lanes 0..15. h double-buffered in global memory
// so only ONE grid barrier per timestep is needed.
__global__ __launch_bounds__(256) void lstm_recur_kernel(
        const float* __restrict__ Whh,   // [4096, 1024]
        const float* __restrict__ xz,    // [T, 4096]
        float* __restrict__ hseq,        // [T, 1024]
        float* __restrict__ hbuf,        // [2, 1024] global h double-buffer
        unsigned* bar, int T) {
    __shared__ float sh[HDIM];
    __shared__ float zs[64];

    const int tid  = threadIdx.x;
    const int q    = tid >> 2;        // quad id 0..63
    const int p    = tid & 3;         // lane in quad
    const int gate = q >> 4;          // 0:i 1:f 2:g 3:o
    const int jl   = q & 15;
    const int j0   = blockIdx.x * 16;
    const int row  = gate * HDIM + j0 + jl;
    const float* wrow = Whh + (size_t)row * HDIM + p * 256;

    float c = 0.0f;                   // cell state (lanes 0..15 use it)

    // h_{-1} = 0 in both buffers
    if (tid < 16) {
        hbuf[j0 + tid]        = 0.0f;
        hbuf[HDIM + j0 + tid] = 0.0f;
    }
    __threadfence();
    grid_barrier(bar, gridDim.x);

    for (int t = 0; t < T; ++t) {
        // stage h_{t-1} (buffer t&1) into LDS: 256 lanes x float4
        ((float4*)sh)[tid] = ((const float4*)(hbuf + (t & 1) * HDIM))[tid];
        __syncthreads();

        // partial dot: 256 elements per lane
        const float* hp = sh + p * 256;
        float acc = 0.0f;
        #pragma unroll 8
        for (int k = 0; k < 256; k += 4) {
            float4 w  = *(const float4*)(wrow + k);
            float4 h4 = *(const float4*)(hp + k);
            acc = fmaf(w.x, h4.x, acc);
            acc = fmaf(w.y, h4.y, acc);
            acc = fmaf(w.z, h4.z, acc);
            acc = fmaf(w.w, h4.w, acc);
        }
        acc += __shfl_xor(acc, 1, 4);
        acc += __shfl_xor(acc, 2, 4);
        if (p == 0)
            zs[gate * 16 + jl] = acc + xz[(size_t)t * G4H + row];
        __syncthreads();

        if (tid < 16) {
            const float zi = zs[tid];
            const float zf = zs[16 + tid];
            const float zg = zs[32 + tid];
            const float zo = zs[48 + tid];
            c = sigmoidf_(zf) * c + sigmoidf_(zi) * tanhf(zg);
            const float h = sigmoidf_(zo) * tanhf(c);
            hbuf[((t + 1) & 1) * HDIM + j0 + tid] = h;   // write buffer (t+1)&1
            hseq[(size_t)t * HDIM + j0 + tid]     = h;
        }
        __threadfence();
        grid_barrier(bar, gridDim.x);   // h_t visible to all before next stage
    }
}

// ------------------------------------------------------------------ FC
__global__ void fc_kernel(const float* __restrict__ h,
                          const float* __restrict__ W,
                          const float* __restrict__ b,
                          float* __restrict__ out, int total) {
    int idx = blockIdx.x * blockDim.x + threadIdx.x;
    if (idx >= total) return;
    int t = idx / 10, o = idx % 10;
    const float* hr = h + (size_t)t * HDIM;
    const float* wr = W + (size_t)o * HDIM;
    float acc = b[o];
    #pragma unroll 4
    for (int k = 0; k < HDIM; k += 4) {
        float4 hv = *(const float4*)(hr + k);
        float4 wv = *(const float4*)(wr + k);
        acc = fmaf(hv.x, wv.x, acc);
        acc = fmaf(hv.y, wv.y, acc);
        acc = fmaf(hv.z, wv.z, acc);
        acc = fmaf(hv.w, wv.w, acc);
    }
    out[idx] = acc;
}

// --------------------------------------------------------------- launcher
extern "C" void kernel_launch(void* const* d_in, const int* in_sizes, int n_in,
                              void* d_out, int out_size, void* d_ws, size_t ws_size,
                              hipStream_t stream) {
    const float* x        = (const float*)d_in[0];  // [4096, 1, 256]
    const float* Wih0     = (const float*)d_in[1];  // [4096, 256]
    const float* Whh0     = (const float*)d_in[2];  // [4096, 1024]
    const float* b0       = (const float*)d_in[3];  // [4096]
    const float* Wih_rest = (const float*)d_in[4];  // [3, 4096, 1024]
    const float* Whh_rest = (const float*)d_in[5];  // [3, 4096, 1024]
    const float* b_rest   = (const float*)d_in[6];  // [3, 4096]
    const float* fcW      = (const float*)d_in[7];  // [10, 1024]
    const float* fcb      = (const float*)d_in[8];  // [10]
    float* out = (float*)d_out;                      // [4096, 1, 10]

    // Workspace layout (~96 MB): barrier | h double-buffer | xz | hA | hB
    char* ws = (char*)d_ws;
    unsigned* bar = (unsigned*)ws;                               // 256 B
    float* hbuf   = (float*)(ws + 256);                          // 8 KB
    float* xz     = (float*)(ws + 256 + 8192);                   // 64 MB
    float* hA     = xz + (size_t)T_SEQ * G4H;                    // 16 MB
    float* hB     = hA + (size_t)T_SEQ * HDIM;                   // 16 MB

    init_barrier_kernel<<<1, 32, 0, stream>>>(bar);

    const dim3 ggrid(G4H / 16, (T_SEQ / 16) / 8);  // (256, 32), 8 waves/block

    // Layer 0 (Din = 256)
    gemm_xz_kernel<<<ggrid, 256, 0, stream>>>(x, Wih0, b0, xz, 256);
    lstm_recur_kernel<<<RBLOCKS, 256, 0, stream>>>(Whh0, xz, hA, hbuf, bar, T_SEQ);

    // Layers 1..3 (Din = 1024), ping-pong hA/hB
    float* hprev = hA;
    float* hnext = hB;
    for (int l = 0; l < 3; ++l) {
        const float* Wih = Wih_rest + (size_t)l * G4H * HDIM;
        const float* Whh = Whh_rest + (size_t)l * G4H * HDIM;
        const float* bb  = b_rest   + (size_t)l * G4H;
        gemm_xz_kernel<<<ggrid, 256, 0, stream>>>(hprev, Wih, bb, xz, HDIM);
        lstm_recur_kernel<<<RBLOCKS, 256, 0, stream>>>(Whh, xz, hnext, hbuf, bar, T_SEQ);
        float* tmp = hprev; hprev = hnext; hnext = tmp;
    }

    // Final FC
    const int total = T_SEQ * 10;
    fc_kernel<<<(total + 255) / 256, 256, 0, stream>>>(hprev, fcW, fcb, out, total);
}